// RegionLoss_13443247636825
// MI455X (gfx1250) — compile-verified
//
#include <hip/hip_runtime.h>
#include <stdint.h>
#include <math.h>

// ---------------- problem constants (from reference) ----------------
#define NBATCH 64
#define NA 5
#define NC 20
#define GG 26
#define NT 256
#define CHN (NC + 5)                 // 25
#define PLANE (GG * GG)              // 676
#define NCELL (NBATCH * NA * PLANE)  // 216320 == 845 * 256 exactly
#define XBSTR (NA * CHN * PLANE)     // batch stride in x
#define THRESH 0.6f
#define EPSI 1e-16f
#define SEEN_WARMUP 12800
#define BLK 256
#define OUT_FLOATS_PER_BLK (BLK * CHN)                // 6400 floats = 25600 B
#define OUT_CHUNKS_PER_BLK (OUT_FLOATS_PER_BLK / 4)   // 1600 b128 chunks

__constant__ float c_aw[NA] = {1.08f, 3.42f, 6.63f, 9.42f, 16.62f};
__constant__ float c_ah[NA] = {1.19f, 4.41f, 11.38f, 5.11f, 10.52f};

__device__ __forceinline__ float sigm(float v) { return 1.0f / (1.0f + __expf(-v)); }

// ---- CDNA5 async LDS<->global helpers (per-lane, tracked with ASYNCcnt) ----
__device__ __forceinline__ void async_load_b128(unsigned lds_off, const void* gptr) {
  asm volatile("global_load_async_to_lds_b128 %0, %1, off"
               :: "v"(lds_off), "v"((unsigned long long)(uintptr_t)gptr) : "memory");
}
__device__ __forceinline__ void async_store_b128(void* gptr, unsigned lds_off) {
  asm volatile("global_store_async_from_lds_b128 %0, %1, off"
               :: "v"((unsigned long long)(uintptr_t)gptr), "v"(lds_off) : "memory");
}
__device__ __forceinline__ void wait_async0() {
#if defined(__has_builtin) && __has_builtin(__builtin_amdgcn_s_wait_asynccnt)
  __builtin_amdgcn_s_wait_asynccnt(0);
#else
  asm volatile("s_wait_asynccnt 0" ::: "memory");
#endif
}

// accumulator slots: 0=loss_coord 1=S_obj 2=S_noobj 3=loss_cls 4=n_obj 5=n_noobj

// ---------------- kernel 1: workspace init (deterministic each call) ---------
__global__ void rl_init(int* __restrict__ cellTarget, int* __restrict__ clsMask,
                        float* __restrict__ acc) {
  int i = blockIdx.x * blockDim.x + threadIdx.x;
  if (i < NCELL) { cellTarget[i] = -1; clsMask[i] = 0; }
  if (i < 8) acc[i] = 0.0f;
}

// ---------------- kernel 2: per-target preprocessing -------------------------
__global__ void rl_prep(const float* __restrict__ targets,
                        float4* __restrict__ tbox, float* __restrict__ tdata,
                        int* __restrict__ cellTarget, int* __restrict__ clsMask) {
  int t = threadIdx.x;  // one block of NT threads
  const float* tg = targets + t * 6;
  int b = (int)tg[0];
  int label = (int)tg[1];
  float w0 = tg[4], h0 = tg[5];
  float gx = tg[2] * GG, gy = tg[3] * GG, gw = w0 * GG, gh = h0 * GG;

  int best = 0; float bestv = -1.0f;
#pragma unroll
  for (int a = 0; a < NA; ++a) {               // argmax: first max wins
    float aw = c_aw[a], ah = c_ah[a];
    float inter = fminf(aw, gw) * fminf(ah, gh);
    float iou = inter / (aw * ah + gw * gh - inter);
    if (iou > bestv) { bestv = iou; best = a; }
  }
  int gi = (int)floorf(gx), gj = (int)floorf(gy);
  int cell = ((b * NA + best) * GG + gj) * GG + gi;

  atomicMax(&cellTarget[cell], t);             // deterministic duplicate winner
  atomicOr(&clsMask[cell], 1 << label);        // tcls scatter sets every label bit

  tbox[t] = make_float4(gx, gy, gw, gh);
  float* td = tdata + t * 8;
  td[0] = gx - floorf(gx);
  td[1] = gy - floorf(gy);
  td[2] = __logf(gw / c_aw[best]);
  td[3] = __logf(gh / c_ah[best]);
  td[4] = 2.0f - w0 * h0;                      // COORD_SCALE == 1
}

// ---------------- kernel 3: main per-cell kernel -----------------------------
__global__ __launch_bounds__(BLK) void rl_main(
    const float* __restrict__ x, const float4* __restrict__ tbox,
    const float* __restrict__ tdata, const int* __restrict__ cellTarget,
    const int* __restrict__ clsMask, const int* __restrict__ seen_p,
    float* __restrict__ out, float* __restrict__ acc) {
  __shared__ float4 shTb[NT];                  // 4 KB target boxes
  __shared__ float shOut[OUT_FLOATS_PER_BLK];  // 25.6 KB staged output rows

  // --- kick off async global->LDS broadcast of the 4KB target-box table;
  //     do NOT wait yet: it is consumed only by the IoU loop far below ---
  async_load_b128((unsigned)(uintptr_t)(&shTb[threadIdx.x]),
                  (const void*)(tbox + threadIdx.x));

  int cell = blockIdx.x * blockDim.x + threadIdx.x;  // NCELL == grid*block exact
  int gi = cell % GG;
  int gj = (cell / GG) % GG;
  int a  = (cell / PLANE) % NA;
  int b  = cell / (PLANE * NA);

  // 25 channel loads, lane-contiguous along gi (coalesced 128B/wave each);
  // these overlap the async table load above
  const float* xb = x + (size_t)b * XBSTR + (size_t)a * CHN * PLANE + gj * GG + gi;
  float ch[CHN];
#pragma unroll
  for (int c = 0; c < CHN; ++c) ch[c] = xb[c * PLANE];

  float pxv = sigm(ch[0]);
  float pyv = sigm(ch[1]);
  float pwv = ch[2];
  float phv = ch[3];
  float pconf = sigm(ch[4]);
  float pcls[NC];
#pragma unroll
  for (int c = 0; c < NC; ++c) pcls[c] = sigm(ch[5 + c]);

  float bx = pxv + (float)gi;
  float by = pyv + (float)gj;
  float bw = __expf(pwv) * c_aw[a];
  float bh = __expf(phv) * c_ah[a];

  // ---- stage output row in LDS (out offset == cell*CHN, block-contiguous) ----
  {
    const float invG = 1.0f / (float)GG;
    float* o = &shOut[threadIdx.x * CHN];      // stride-25 words: conflict-free
    o[0] = bx * invG; o[1] = by * invG; o[2] = bw * invG; o[3] = bh * invG;
    o[4] = pconf;
#pragma unroll
    for (int c = 0; c < NC; ++c) o[5 + c] = pcls[c];
  }

  // precise wait: only the shTb async load is outstanding at this point
  wait_async0();
  __syncthreads();   // single barrier publishes BOTH shTb and shOut

  // ---- drain staged rows with coalesced async LDS->global b128 stores ----
  // issued BEFORE the IoU loop so the copy overlaps the VALU-heavy phase;
  // S_ENDPGM's implicit wait-idle drains ASYNCcnt.
  {
    float* outBase = out + (size_t)blockIdx.x * OUT_FLOATS_PER_BLK;
    unsigned shBase = (unsigned)(uintptr_t)(&shOut[0]);
    for (int c = threadIdx.x; c < OUT_CHUNKS_PER_BLK; c += BLK)
      async_store_b128((void*)(outBase + (size_t)c * 4), shBase + (unsigned)c * 16u);
  }

  // ---- max IoU vs all NT targets (LDS broadcast loop: hot path) ----
  float b1x1 = bx - bw * 0.5f, b1x2 = bx + bw * 0.5f;
  float b1y1 = by - bh * 0.5f, b1y2 = by + bh * 0.5f;
  float area1 = bw * bh;
  float maxiou = 0.0f;
#pragma unroll 4
  for (int t = 0; t < NT; ++t) {
    float4 tb = shTb[t];
    float hw = tb.z * 0.5f, hh = tb.w * 0.5f;
    float iw = fmaxf(fminf(b1x2, tb.x + hw) - fmaxf(b1x1, tb.x - hw), 0.0f);
    float ih = fmaxf(fminf(b1y2, tb.y + hh) - fmaxf(b1y1, tb.y - hh), 0.0f);
    float inter = iw * ih;
    maxiou = fmaxf(maxiou, inter / (area1 + tb.z * tb.w - inter + EPSI));
  }

  int seen = seen_p[0];
  float coord0 = (seen < SEEN_WARMUP) ? 1.0f : 0.0f;
  float txy0   = (seen < SEEN_WARMUP) ? 0.5f : 0.0f;

  int sel = cellTarget[cell];
  float lc = 0.f, sobj = 0.f, snoobj = 0.f, lcls = 0.f, cobj = 0.f, cnoobj = 0.f;
  if (sel >= 0) {
    const float* td = tdata + sel * 8;
    float dx = pxv - td[0], dy = pyv - td[1];
    float dw = pwv - td[2], dh = phv - td[3];
    lc = td[4] * (dx * dx + dy * dy + dw * dw + dh * dh);

    float4 tb = shTb[sel];                     // iou_sel (RESCORE -> tconf)
    float hw = tb.z * 0.5f, hh = tb.w * 0.5f;
    float iw = fmaxf(fminf(b1x2, tb.x + hw) - fmaxf(b1x1, tb.x - hw), 0.0f);
    float ih = fmaxf(fminf(b1y2, tb.y + hh) - fmaxf(b1y1, tb.y - hh), 0.0f);
    float inter = iw * ih;
    float tconf = inter / (area1 + tb.z * tb.w - inter + EPSI);
    float dc = pconf - tconf;
    sobj = dc * dc;
    cobj = 1.0f;

    int mask = clsMask[cell];
#pragma unroll
    for (int c = 0; c < NC; ++c) {
      float p = fminf(fmaxf(pcls[c], 1e-7f), 1.0f - 1e-7f);
      float tcv = (float)((mask >> c) & 1);
      lcls -= tcv * __logf(p) + (1.0f - tcv) * __logf(1.0f - p);
    }
  } else {
    if (coord0 != 0.0f) {                      // seen < warmup path
      float dx = pxv - txy0, dy = pyv - txy0;
      lc = coord0 * (dx * dx + dy * dy + pwv * pwv + phv * phv);
    }
    if (maxiou <= THRESH) { snoobj = pconf * pconf; cnoobj = 1.0f; }
  }

  // ---- wave32 shuffle reduction, one atomic per wave per slot ----
  float vals[6] = {lc, sobj, snoobj, lcls, cobj, cnoobj};
#pragma unroll
  for (int k = 0; k < 6; ++k) {
    float v = vals[k];
#pragma unroll
    for (int off = 16; off > 0; off >>= 1) v += __shfl_down(v, off, 32);
    if ((threadIdx.x & 31) == 0) atomicAdd(&acc[k], v);
  }
}

// ---------------- kernel 4: finalize total loss ------------------------------
__global__ void rl_finalize(const float* __restrict__ acc, float* __restrict__ loss) {
  // total = Lcoord + (n_noobj/n_obj)*S_obj + 1.0*S_noobj + Lcls
  loss[0] = acc[0] + (acc[5] / acc[4]) * acc[1] + acc[2] + acc[3];
}

// ---------------- host launcher ---------------------------------------------
extern "C" void kernel_launch(void* const* d_in, const int* in_sizes, int n_in,
                              void* d_out, int out_size, void* d_ws, size_t ws_size,
                              hipStream_t stream) {
  const float* x       = (const float*)d_in[0];
  const float* targets = (const float*)d_in[1];
  const int*   seen    = (const int*)d_in[2];
  float* out = (float*)d_out;

  char* ws = (char*)d_ws;
  float*  acc        = (float*)ws;                                  // 32 B
  float4* tbox       = (float4*)(ws + 64);                          // 4 KB
  float*  tdata      = (float*)(ws + 64 + 4096);                    // 8 KB
  int*    cellTarget = (int*)(ws + 16384);                          // 845 KB
  int*    clsMask    = (int*)(ws + 16384 + (size_t)NCELL * 4);      // 845 KB

  rl_init<<<(NCELL + 255) / 256, 256, 0, stream>>>(cellTarget, clsMask, acc);
  rl_prep<<<1, NT, 0, stream>>>(targets, tbox, tdata, cellTarget, clsMask);
  rl_main<<<NCELL / BLK, BLK, 0, stream>>>(x, tbox, tdata, cellTarget, clsMask,
                                           seen, out, acc);
  rl_finalize<<<1, 1, 0, stream>>>(acc, out + (out_size - 1));
}